// DifferentiableSTA_91233695301870
// MI455X (gfx1250) — compile-verified
//
#include <hip/hip_runtime.h>
#include <cstdint>
#include <cstddef>

// ---------------------------------------------------------------------------
// Differentiable STA forward sweep for MI455X (gfx1250).
//   W=2048 nodes/level, L=128 levels, K=16 fanin, M_EP=2048 endpoints.
// Roofline: ~152 MB streamed once -> ~6.5us at 23.3 TB/s; compute is trivial
// (~0.13 GFLOP + 17M v_exp_f32). The bottleneck is the 127-deep level
// dependency chain, so: ONE persistent kernel (128 blocks x 256 threads,
// trivially co-resident) with a device-scope barrier per level instead of 127
// kernel launches. Previous-level arrivals (16 KB) are staged into LDS with
// gfx1250 async global->LDS loads, issued BEFORE the level's main per-edge
// HBM stream so staging overlaps it; random gathers then hit LDS. One lane
// per incoming edge; masked log-sum-exp via 16-wide wave32 shuffles.
// WMMA is inapplicable here (sparse gathers + LSE, no dense MACs).
// ---------------------------------------------------------------------------

#define STA_W   2048
#define STA_L   128
#define STA_K   16
#define STA_MEP 2048
#define NEG_INF (-1e30f)
#define TAU_F   0.07f

#if __has_builtin(__builtin_amdgcn_global_load_async_to_lds_b128)
#define HAVE_ASYNC_LDS 1
#else
#define HAVE_ASYNC_LDS 0
#endif

// Pointer element type required by the async-to-LDS builtins: int32x4.
typedef int v4i_vec __attribute__((vector_size(16)));
typedef __attribute__((address_space(1))) v4i_vec GlobV4;  // global (AS1)
typedef __attribute__((address_space(3))) v4i_vec LdsV4;   // LDS (AS3)

// ---------------------------------------------------------------------------
// Init: copy level-0 arrivals into at_all, zero the barrier counters.
// ---------------------------------------------------------------------------
__global__ void sta_init_kernel(const float* __restrict__ input_arrival,
                                float* __restrict__ at_all,
                                unsigned* __restrict__ counters) {
  int i = blockIdx.x * blockDim.x + threadIdx.x;
  if (i < STA_W * 2) at_all[i] = input_arrival[i];
  if (i < STA_L) counters[i] = 0u;
}

// ---------------------------------------------------------------------------
// Persistent levelized sweep. Grid: 128 blocks x 256 threads.
// Block b handles nodes [16*b, 16*b+16) of every level; lane e of each
// 16-lane group handles edge e of its node.
// ---------------------------------------------------------------------------
__global__ __launch_bounds__(256) void sta_sweep_kernel(
    const float* __restrict__ d_hat,        // [E,4]
    const float* __restrict__ sta_mask,     // [E,4]
    const int*   __restrict__ edge_src,     // [E]
    const float* __restrict__ input_arrival,// [N,2]
    float* __restrict__ at_all,             // [N,2] (output region)
    unsigned* __restrict__ counters)        // [L] barrier counters (zeroed)
{
  __shared__ float prev[STA_W * 2];         // 16 KB: previous level arrivals
  const int tid  = threadIdx.x;
  const int lane = tid & (STA_K - 1);
  const int node = blockIdx.x * 16 + (tid >> 4);
  const float inv_tau = 1.0f / TAU_F;
  const float4* __restrict__ d4p = reinterpret_cast<const float4*>(d_hat);
  const float4* __restrict__ m4p = reinterpret_cast<const float4*>(sta_mask);

  for (int level = 1; level < STA_L; ++level) {
    // ---- kick off async staging of previous level arrivals (16 KB) ----
    const char* gsrc =
        reinterpret_cast<const char*>(at_all + (size_t)(level - 1) * STA_W * 2);
#if HAVE_ASYNC_LDS
    #pragma unroll
    for (int k = 0; k < 4; ++k) {
      unsigned off = (unsigned)(tid + k * 256) * 16u;   // 1024 x 16B chunks
      __builtin_amdgcn_global_load_async_to_lds_b128(
          (GlobV4*)(gsrc + off), (LdsV4*)((char*)prev + off), 0, 0);
    }
#endif

    // ---- issue this level's per-edge HBM stream (independent of LDS) ----
    const size_t e =
        ((size_t)(level - 1) * STA_W + (size_t)node) * STA_K + (size_t)lane;
    const int src  = edge_src[e] & (STA_W - 1);   // relative to previous level
    const float4 d4 = d4p[e];
    const float4 m4 = m4p[e];

    // ---- now drain the async staging and make LDS visible block-wide ----
#if HAVE_ASYNC_LDS
#if __has_builtin(__builtin_amdgcn_s_wait_asynccnt)
    __builtin_amdgcn_s_wait_asynccnt(0);
#else
    asm volatile("s_wait_asynccnt 0x0" ::: "memory");
#endif
#else
    {
      const float4* g4 = reinterpret_cast<const float4*>(gsrc);
      float4* l4 = reinterpret_cast<float4*>(prev);
      for (int k = tid; k < (STA_W * 2) / 4; k += 256) l4[k] = g4[k];
    }
#endif
    __syncthreads();

    // ---- gather + candidates ----
    const float ur = prev[src * 2 + 0];
    const float uf = prev[src * 2 + 1];
    const bool rr = ur > NEG_INF;                 // REACH_THR == NEG_INF in f32
    const bool rf = uf > NEG_INF;
    const float c0 = ur + d4.x; const bool v0 = (m4.x > 0.5f) && rr; // r->r
    const float c1 = ur + d4.y; const bool v1 = (m4.y > 0.5f) && rr; // r->f
    const float c2 = uf + d4.z; const bool v2 = (m4.z > 0.5f) && rf; // f->r
    const float c3 = uf + d4.w; const bool v3 = (m4.w > 0.5f) && rf; // f->f

    // ---- masked LSE over the node's 32 candidates (16-lane reductions) ----
    float mr = fmaxf(v0 ? c0 : NEG_INF, v2 ? c2 : NEG_INF);
    float mf = fmaxf(v1 ? c1 : NEG_INF, v3 ? c3 : NEG_INF);
    #pragma unroll
    for (int o = 8; o > 0; o >>= 1) {
      mr = fmaxf(mr, __shfl_xor(mr, o, 16));
      mf = fmaxf(mf, __shfl_xor(mf, o, 16));
    }
    float sr = (v0 ? __expf((c0 - mr) * inv_tau) : 0.0f)
             + (v2 ? __expf((c2 - mr) * inv_tau) : 0.0f);
    float sf = (v1 ? __expf((c1 - mf) * inv_tau) : 0.0f)
             + (v3 ? __expf((c3 - mf) * inv_tau) : 0.0f);
    #pragma unroll
    for (int o = 8; o > 0; o >>= 1) {
      sr += __shfl_xor(sr, o, 16);
      sf += __shfl_xor(sf, o, 16);
    }

    if (lane == 0) {
      const size_t nid = (size_t)level * STA_W + (size_t)node;
      const float old_r = input_arrival[nid * 2 + 0];
      const float old_f = input_arrival[nid * 2 + 1];
      // any valid candidate <=> reduced max rose above NEG_INF
      const float new_r =
          (mr > NEG_INF) ? (mr + TAU_F * __logf(sr + 1e-30f)) : old_r;
      const float new_f =
          (mf > NEG_INF) ? (mf + TAU_F * __logf(sf + 1e-30f)) : old_f;
      at_all[nid * 2 + 0] = new_r;
      at_all[nid * 2 + 1] = new_f;
    }

    // pull next level's edge payloads toward L2/L0 while we sit in the barrier
    if (level + 1 < STA_L) {
      const size_t en = e + (size_t)STA_W * STA_K;
      __builtin_prefetch(&edge_src[en], 0, 0);
      __builtin_prefetch(&d4p[en], 0, 0);
      __builtin_prefetch(&m4p[en], 0, 0);
    }

    // ---- device-wide level barrier (single-use counter per level) ----
    __threadfence();
    __syncthreads();
    if (tid == 0) {
      __hip_atomic_fetch_add(&counters[level], 1u, __ATOMIC_ACQ_REL,
                             __HIP_MEMORY_SCOPE_AGENT);
      while (__hip_atomic_load(&counters[level], __ATOMIC_ACQUIRE,
                               __HIP_MEMORY_SCOPE_AGENT) < (unsigned)gridDim.x) {
        __builtin_amdgcn_s_sleep(2);
      }
    }
    __syncthreads();
    __threadfence();   // acquire: other blocks' level-l arrivals now visible
  }
}

// ---------------------------------------------------------------------------
// Endpoint gather + slack.
// ---------------------------------------------------------------------------
__global__ void sta_endpoint_kernel(const float* __restrict__ at_all,
                                    const int* __restrict__ endpoint_ids,
                                    const float* __restrict__ rat_true,
                                    float* __restrict__ at_ep_safe,
                                    float* __restrict__ slack) {
  int i = blockIdx.x * blockDim.x + threadIdx.x;
  if (i < STA_MEP * 2) {
    int idx = i >> 1, c = i & 1;
    int ep = endpoint_ids[idx];
    float at = at_all[(size_t)ep * 2 + c];
    float safe = (at > NEG_INF) ? at : 0.0f;
    at_ep_safe[i] = safe;
    slack[i] = rat_true[i] - safe;
  }
}

// ---------------------------------------------------------------------------
extern "C" void kernel_launch(void* const* d_in, const int* in_sizes, int n_in,
                              void* d_out, int out_size, void* d_ws,
                              size_t ws_size, hipStream_t stream) {
  (void)in_sizes; (void)n_in; (void)out_size; (void)ws_size;
  const float* d_hat         = (const float*)d_in[0];   // [E,4]
  const float* sta_mask      = (const float*)d_in[1];   // [E,4]
  const int*   edge_src      = (const int*)d_in[2];     // [E]
  // d_in[3] edge_dst, d_in[4] topo_order: implicit in the levelized layout
  const float* input_arrival = (const float*)d_in[5];   // [N,2]
  const int*   endpoint_ids  = (const int*)d_in[6];     // [M_EP]
  const float* rat_true      = (const float*)d_in[7];   // [M_EP,2]

  float* at_all = (float*)d_out;                        // [N,2]
  float* at_ep  = at_all + (size_t)STA_L * STA_W * 2;   // [M_EP,2]
  float* slack  = at_ep + (size_t)STA_MEP * 2;          // [M_EP,2]
  unsigned* counters = (unsigned*)d_ws;                 // L u32, zeroed per call

  sta_init_kernel<<<(STA_W * 2 + 255) / 256, 256, 0, stream>>>(
      input_arrival, at_all, counters);
  sta_sweep_kernel<<<STA_W / 16, 256, 0, stream>>>(
      d_hat, sta_mask, edge_src, input_arrival, at_all, counters);
  sta_endpoint_kernel<<<(STA_MEP * 2 + 255) / 256, 256, 0, stream>>>(
      at_all, endpoint_ids, rat_true, at_ep, slack);
}